// SquareGNN_6055903888071
// MI455X (gfx1250) — compile-verified
//
#include <hip/hip_runtime.h>
#include <hip/hip_bf16.h>
#include <math.h>

// ---------------------------------------------------------------------------
// GATv2 x2 on gfx1250 (wave32). fp32 WMMA (V_WMMA_F32_16X16X4_F32) for the
// node-feature GEMMs; 3-pass edge softmax with native f32 global atomics.
// Edge gathers vectorized to b128/b64 loads (bandwidth-bound phase).
// ---------------------------------------------------------------------------

typedef __attribute__((ext_vector_type(2))) float v2f;
typedef __attribute__((ext_vector_type(8))) float v8f;

// --------------------------- dual GEMM (WMMA f32) --------------------------
// Computes xl = X @ Wl and xr = X @ Wr for a 16-row M tile per block.
// X: [N,K] row-major, Wl/Wr: [K,HC] row-major, outputs [N,HC].
// Block = 256 threads = 8 waves; wave w handles N-tile (blockIdx.y*8 + w)
// out of 2*HC/16 tiles (first half -> Wl/xl, second half -> Wr/xr).
template <int K, int HC>
__global__ __launch_bounds__(256) void gemm_dual_wmma(
    const float* __restrict__ X,
    const float* __restrict__ Wl, const float* __restrict__ Wr,
    float* __restrict__ xl, float* __restrict__ xr)
{
    constexpr int LDSROW = K + 4;                 // pad: conflict-free frag reads
    __shared__ float As[16 * LDSROW];

    const int m0 = blockIdx.x * 16;
    // Cooperative load of the 16xK A tile into LDS (vectorized b128).
    {
        const float4* Xv = (const float4*)(X + (size_t)m0 * K);
        for (int i = threadIdx.x; i < 16 * K / 4; i += 256) {
            const int r = i / (K / 4), c4 = i % (K / 4);
            const float4 v = Xv[(size_t)r * (K / 4) + c4];
            float* dst = &As[r * LDSROW + c4 * 4];
            dst[0] = v.x; dst[1] = v.y; dst[2] = v.z; dst[3] = v.w;
        }
    }
    __syncthreads();

    const int wave = threadIdx.x >> 5;
    const int lane = threadIdx.x & 31;
    const int ntile = blockIdx.y * 8 + wave;
    constexpr int NT_HALF = HC / 16;

    const float* W;
    float* Out;
    int n0;
    if (ntile < NT_HALF) { W = Wl; Out = xl; n0 = ntile * 16; }
    else                 { W = Wr; Out = xr; n0 = (ntile - NT_HALF) * 16; }

    // f32 16x16x4 fragment layout (ISA 7.12.2):
    //   A: v_j = A[M = lane&15, K = 2*(lane>>4)+j]
    //   B: v_j = B[K = 2*(lane>>4)+j, N = lane&15]
    //   C/D: vgpr r, lane L -> row r + 8*(L>>4), col L&15
    const int row   = lane & 15;
    const int khalf = (lane >> 4) * 2;
    const int ncol  = n0 + (lane & 15);

    v8f acc = {};
#pragma unroll 4
    for (int kk = 0; kk < K; kk += 4) {
        const int kb = kk + khalf;
        v2f a, b;
        a.x = As[row * LDSROW + kb];
        a.y = As[row * LDSROW + kb + 1];
        b.x = W[(size_t)kb * HC + ncol];
        b.y = W[(size_t)(kb + 1) * HC + ncol];
        acc = __builtin_amdgcn_wmma_f32_16x16x4_f32(
            /*neg_a=*/false, a, /*neg_b=*/false, b,
            /*c_mod=*/(short)0, acc, /*reuse_a=*/false, /*reuse_b=*/false);
    }

    const int rbase = (lane >> 4) * 8;
#pragma unroll
    for (int r = 0; r < 8; ++r)
        Out[(size_t)(m0 + rbase + r) * HC + ncol] = acc[r];
}

// ------------------------------ edge helpers -------------------------------
__device__ __forceinline__ void edge_ends(const long long* __restrict__ ei,
                                          int E, int w, int& s, int& d)
{
    if (w < E) { s = (int)ei[w]; d = (int)ei[(size_t)E + w]; }
    else       { s = d = w - E; }      // appended self-loop
}

__device__ __forceinline__ int edge_dst(const long long* __restrict__ ei,
                                        int E, int w)
{
    return (w < E) ? (int)ei[(size_t)E + w] : (w - E);
}

__device__ __forceinline__ float lrelu(float m) { return m > 0.f ? m : 0.2f * m; }

// Pass 1: per-edge GATv2 logits e = sum_c leakyrelu(xl[s]+xr[d]) * att,
// plus atomic segment-max into emax[dst]. One wave per edge.
template <int H, int C>
__global__ __launch_bounds__(256) void edge_logits(
    const long long* __restrict__ ei, int E, int Etot,
    const float* __restrict__ xl, const float* __restrict__ xr,
    const float* __restrict__ att,
    float* __restrict__ e, float* __restrict__ emax)
{
    constexpr int HC  = H * C;
    constexpr int CPL = HC / 32;      // channels per lane
    constexpr int LPH = 32 / H;       // lanes per head
    const int w = blockIdx.x * 8 + (threadIdx.x >> 5);
    const int lane = threadIdx.x & 31;
    if (w >= Etot) return;

    int s, d;
    edge_ends(ei, E, w, s, d);

    const int c0 = lane * CPL;
    const int h  = c0 / C;
    const float* pl = xl + (size_t)s * HC + c0;
    const float* pr = xr + (size_t)d * HC + c0;
    const float* pa = att + c0;

    float p = 0.f;
    if constexpr (CPL % 4 == 0) {                 // b128 gathers
        const float4* l4 = (const float4*)pl;
        const float4* r4 = (const float4*)pr;
        const float4* a4 = (const float4*)pa;
#pragma unroll
        for (int j = 0; j < CPL / 4; ++j) {
            const float4 lv = l4[j], rv = r4[j], av = a4[j];
            p += lrelu(lv.x + rv.x) * av.x;
            p += lrelu(lv.y + rv.y) * av.y;
            p += lrelu(lv.z + rv.z) * av.z;
            p += lrelu(lv.w + rv.w) * av.w;
        }
    } else {                                      // CPL == 2: b64 gathers
        const float2 lv = *(const float2*)pl;
        const float2 rv = *(const float2*)pr;
        const float2 av = *(const float2*)pa;
        p += lrelu(lv.x + rv.x) * av.x;
        p += lrelu(lv.y + rv.y) * av.y;
    }

#pragma unroll
    for (int off = LPH >> 1; off > 0; off >>= 1)
        p += __shfl_xor(p, off, 32);

    if ((lane & (LPH - 1)) == 0) {
        e[(size_t)w * H + h] = p;
        atomicMax(&emax[(size_t)d * H + h], p);    // GLOBAL_ATOMIC_MAX_NUM_F32
    }
}

// Pass 2: ee = exp(e - emax[dst]); atomic segment-sum into denom[dst].
template <int H>
__global__ __launch_bounds__(256) void edge_exp(
    const long long* __restrict__ ei, int E, int Etot,
    float* __restrict__ e, const float* __restrict__ emax,
    float* __restrict__ denom)
{
    const int t = blockIdx.x * blockDim.x + threadIdx.x;
    if (t >= Etot * H) return;
    const int w = t / H, h = t - w * H;
    const int d = edge_dst(ei, E, w);
    const float v = expf(e[t] - emax[(size_t)d * H + h]);
    e[t] = v;
    atomicAdd(&denom[(size_t)d * H + h], v);
}

// Pass 3: alpha = ee/denom[dst]; agg[dst] += alpha * xl[src]. One wave/edge.
template <int H, int C>
__global__ __launch_bounds__(256) void edge_scatter(
    const long long* __restrict__ ei, int E, int Etot,
    const float* __restrict__ xl, const float* __restrict__ e,
    const float* __restrict__ denom, float* __restrict__ agg)
{
    constexpr int HC  = H * C;
    constexpr int CPL = HC / 32;
    const int w = blockIdx.x * 8 + (threadIdx.x >> 5);
    const int lane = threadIdx.x & 31;
    if (w >= Etot) return;

    int s, d;
    edge_ends(ei, E, w, s, d);

    const int c0 = lane * CPL;
    const int h  = c0 / C;
    const float alpha = e[(size_t)w * H + h] / denom[(size_t)d * H + h];
    const float* pl = xl + (size_t)s * HC + c0;
    float* po = agg + (size_t)d * HC + c0;

    if constexpr (CPL % 4 == 0) {                 // b128 gather, scalar atomics
        const float4* l4 = (const float4*)pl;
#pragma unroll
        for (int j = 0; j < CPL / 4; ++j) {
            const float4 lv = l4[j];
            atomicAdd(&po[j * 4 + 0], alpha * lv.x);   // GLOBAL_ATOMIC_ADD_F32
            atomicAdd(&po[j * 4 + 1], alpha * lv.y);
            atomicAdd(&po[j * 4 + 2], alpha * lv.z);
            atomicAdd(&po[j * 4 + 3], alpha * lv.w);
        }
    } else {                                      // CPL == 2
        const float2 lv = *(const float2*)pl;
        atomicAdd(&po[0], alpha * lv.x);
        atomicAdd(&po[1], alpha * lv.y);
    }
}

// Init: emax=-inf, denom=0, agg = bias broadcast.
__global__ __launch_bounds__(256) void init_layer(
    float* __restrict__ emax, float* __restrict__ denom,
    float* __restrict__ agg, const float* __restrict__ bias,
    int N, int HC, int H)
{
    const int i = blockIdx.x * blockDim.x + threadIdx.x;
    if (i < N * HC) agg[i] = bias[i % HC];
    if (i < N * H)  { emax[i] = -3.402823466e38f; denom[i] = 0.f; }
}

// Exact-erf GELU, in place.
__global__ __launch_bounds__(256) void gelu_kernel(float* __restrict__ h, int n)
{
    const int i = blockIdx.x * blockDim.x + threadIdx.x;
    if (i >= n) return;
    const float x = h[i];
    h[i] = 0.5f * x * (1.f + erff(x * 0.70710678118654752440f));
}

// ------------------------------- launcher ----------------------------------
extern "C" void kernel_launch(void* const* d_in, const int* in_sizes, int n_in,
                              void* d_out, int out_size, void* d_ws, size_t ws_size,
                              hipStream_t stream)
{
    const float* x    = (const float*)d_in[0];   // [N,64]
    const float* Wl1  = (const float*)d_in[1];   // [64,256]
    const float* Wr1  = (const float*)d_in[2];   // [64,256]
    const float* att1 = (const float*)d_in[3];   // [4,64]
    const float* b1   = (const float*)d_in[4];   // [256]
    const float* Wl2  = (const float*)d_in[5];   // [256,64]
    const float* Wr2  = (const float*)d_in[6];   // [256,64]
    const float* att2 = (const float*)d_in[7];   // [1,64]
    const float* b2   = (const float*)d_in[8];   // [64]
    const long long* ei = (const long long*)d_in[9];  // [2,E] int64

    const int N    = in_sizes[0] / 64;
    const int E    = in_sizes[9] / 2;
    const int Etot = E + N;                      // with self-loops

    // Workspace layout (floats). Layer-2 buffers alias layer-1 (safe ordering).
    float* ws   = (float*)d_ws;
    size_t off  = 0;
    float* xl1  = ws + off; off += (size_t)N * 256;
    float* xr1  = ws + off; off += (size_t)N * 256;
    float* hbuf = ws + off; off += (size_t)N * 256;      // agg1 -> gelu -> h
    float* ebuf = ws + off; off += (size_t)Etot * 4;
    float* emax = ws + off; off += (size_t)N * 4;
    float* den  = ws + off; off += (size_t)N * 4;
    float* xl2 = xl1, *xr2 = xr1, *e2 = ebuf, *emax2 = emax, *den2 = den;
    float* out = (float*)d_out;

    const dim3 blk(256);
    const int edgeBlocks = (Etot * 32 + 255) / 256;      // one wave per edge

    // ---------------- Layer 1 (H=4, C=64, K=64, HC=256) ----------------
    gemm_dual_wmma<64, 256><<<dim3(N / 16, 4), blk, 0, stream>>>(x, Wl1, Wr1, xl1, xr1);
    init_layer<<<(N * 256 + 255) / 256, blk, 0, stream>>>(emax, den, hbuf, b1, N, 256, 4);
    edge_logits<4, 64><<<edgeBlocks, blk, 0, stream>>>(ei, E, Etot, xl1, xr1, att1, ebuf, emax);
    edge_exp<4><<<(Etot * 4 + 255) / 256, blk, 0, stream>>>(ei, E, Etot, ebuf, emax, den);
    edge_scatter<4, 64><<<edgeBlocks, blk, 0, stream>>>(ei, E, Etot, xl1, ebuf, den, hbuf);
    gelu_kernel<<<(N * 256 + 255) / 256, blk, 0, stream>>>(hbuf, N * 256);

    // ---------------- Layer 2 (H=1, C=64, K=256, HC=64) ----------------
    gemm_dual_wmma<256, 64><<<dim3(N / 16, 1), blk, 0, stream>>>(hbuf, Wl2, Wr2, xl2, xr2);
    init_layer<<<(N * 64 + 255) / 256, blk, 0, stream>>>(emax2, den2, out, b2, N, 64, 1);
    edge_logits<1, 64><<<edgeBlocks, blk, 0, stream>>>(ei, E, Etot, xl2, xr2, att2, e2, emax2);
    edge_exp<1><<<(Etot + 255) / 256, blk, 0, stream>>>(ei, E, Etot, e2, emax2, den2);
    edge_scatter<1, 64><<<edgeBlocks, blk, 0, stream>>>(ei, E, Etot, xl2, e2, den2, out);
}